// ConvTranspose3d_58909771431981
// MI455X (gfx1250) — compile-verified
//
#include <hip/hip_runtime.h>

typedef __attribute__((ext_vector_type(2))) float v2f;
typedef __attribute__((ext_vector_type(8))) float v8f;

#define CIN  64
#define COUT 64
#define KV   27
#define LDS_STRIDE 66   // 64 + 2 pad: staggers LDS banks, keeps 8B alignment for float2

// ---------------------------------------------------------------------------
// Kernel 1: fill output table with broadcast bias (this is ~415MB => float4).
// ---------------------------------------------------------------------------
__global__ __launch_bounds__(256) void spconvt_bias_init(
    float* __restrict__ out, const float* __restrict__ bias, int n4)
{
    int t = blockIdx.x * 256 + threadIdx.x;
    if (t < n4) {
        const float4* b4 = (const float4*)bias;    // 16 float4 per 64-col row
        ((float4*)out)[t] = b4[t & 15];
    }
}

// ---------------------------------------------------------------------------
// Kernel 2: per-offset GEMM (fp32 WMMA 16x16x4) + scatter-add via f32 atomics.
// Block = 128 threads (4 waves). Block covers 64 points for one kernel offset.
// ---------------------------------------------------------------------------
__global__ __launch_bounds__(128) void spconvt_gemm_scatter(
    const float* __restrict__ feats,      // [N, 64]
    const float* __restrict__ weight,     // [27, 64(Cout), 64(Cin)]
    const int*   __restrict__ out_index,  // [27, N]
    float*       __restrict__ out,        // [N_OUT, 64]
    int npts)
{
    __shared__ float sA[64 * LDS_STRIDE]; // feats tile   [row][cin]
    __shared__ float sB[64 * LDS_STRIDE]; // weight[k]    [cout][cin]

    const int tid  = threadIdx.x;
    const int lane = tid & 31;
    const int wave = tid >> 5;
    const int k    = blockIdx.y;            // kernel offset 0..26
    const int row0 = blockIdx.x * 64;       // first point of the tile

    // ---- cooperative LDS staging: 64x64 feats tile + 64x64 weight[k] ----
    const float* wk = weight + (size_t)k * COUT * CIN;
    #pragma unroll
    for (int i = 0; i < 8; ++i) {
        int f4 = tid + i * 128;             // which float4 of 1024
        int r  = f4 >> 4;                   // row 0..63
        int c4 = (f4 & 15) << 2;            // col 0,4,...,60
        float4 va = make_float4(0.f, 0.f, 0.f, 0.f);
        int gr = row0 + r;
        if (gr < npts)
            va = *(const float4*)(feats + (size_t)gr * CIN + c4);
        *(float4*)(&sA[r * LDS_STRIDE + c4]) = va;
        float4 vb = *(const float4*)(wk + r * CIN + c4);
        *(float4*)(&sB[r * LDS_STRIDE + c4]) = vb;
    }
    __syncthreads();

    // ---- WMMA: wave owns 16 M-rows x 64 N-cols = four 16x16 f32 tiles ----
    // A frag (16x4 f32): lane&15 = M row; lanes>=16 take K+2; 2 VGPRs = K,K+1.
    const int arow = wave * 16 + (lane & 15);
    const int kHi  = (lane >> 4) << 1;
    const int bcol = lane & 15;             // B: lane&15 = N col, same K split

    v8f acc0 = {}, acc1 = {}, acc2 = {}, acc3 = {};

    #pragma unroll
    for (int kk = 0; kk < 16; ++kk) {       // 16 K-steps of 4 => CIN=64
        int c = (kk << 2) + kHi;
        v2f a  = *(const v2f*)(&sA[arow * LDS_STRIDE + c]);
        v2f b0 = *(const v2f*)(&sB[( 0 + bcol) * LDS_STRIDE + c]);
        v2f b1 = *(const v2f*)(&sB[(16 + bcol) * LDS_STRIDE + c]);
        v2f b2 = *(const v2f*)(&sB[(32 + bcol) * LDS_STRIDE + c]);
        v2f b3 = *(const v2f*)(&sB[(48 + bcol) * LDS_STRIDE + c]);
        acc0 = __builtin_amdgcn_wmma_f32_16x16x4_f32(false, a, false, b0, (short)0, acc0, false, false);
        acc1 = __builtin_amdgcn_wmma_f32_16x16x4_f32(false, a, false, b1, (short)0, acc1, false, false);
        acc2 = __builtin_amdgcn_wmma_f32_16x16x4_f32(false, a, false, b2, (short)0, acc2, false, false);
        acc3 = __builtin_amdgcn_wmma_f32_16x16x4_f32(false, a, false, b3, (short)0, acc3, false, false);
    }

    // ---- scatter-add. D layout: lane&15 = N col; VGPR j = M row j (+8 hi half)
    const int ocol  = lane & 15;
    const int rbase = row0 + wave * 16 + ((lane >> 4) << 3);
    const int* oidx = out_index + (size_t)k * npts;
    #pragma unroll
    for (int j = 0; j < 8; ++j) {
        int n = rbase + j;
        if (n < npts) {
            int orow = oidx[n];
            float* dst = out + (size_t)orow * COUT;
            atomicAdd(dst + ( 0 + ocol), acc0[j]);
            atomicAdd(dst + (16 + ocol), acc1[j]);
            atomicAdd(dst + (32 + ocol), acc2[j]);
            atomicAdd(dst + (48 + ocol), acc3[j]);
        }
    }
}

// ---------------------------------------------------------------------------
extern "C" void kernel_launch(void* const* d_in, const int* in_sizes, int n_in,
                              void* d_out, int out_size, void* d_ws, size_t ws_size,
                              hipStream_t stream)
{
    const float* feats     = (const float*)d_in[0];   // [N, 64] f32
    const float* weight    = (const float*)d_in[1];   // [27, 64, 64] f32
    const float* bias      = (const float*)d_in[2];   // [64] f32
    const int*   out_index = (const int*)d_in[3];     // [27, N] i32
    float*       out       = (float*)d_out;           // [N_OUT, 64] f32

    const int npts = in_sizes[0] / CIN;

    // 1) out = bias (broadcast), vectorized float4
    int n4 = out_size / 4;
    spconvt_bias_init<<<dim3((n4 + 255) / 256), 256, 0, stream>>>(out, bias, n4);

    // 2) per-offset WMMA GEMM + atomic scatter
    dim3 grid((npts + 63) / 64, KV);
    spconvt_gemm_scatter<<<grid, 128, 0, stream>>>(feats, weight, out_index, out, npts);
}